// TaggerGreedyDecoder_37331855737105
// MI455X (gfx1250) — compile-verified
//
#include <hip/hip_runtime.h>
#include <hip/hip_bf16.h>
#include <stdint.h>
#include <stddef.h>

// Viterbi decode (CRF) for B=1024, T=512, K=64 on gfx1250 (MI455X).
// Strategy: max-only forward (2 VALU/elem), store alpha history to d_ws
// (needs B*(T+1)*K*4 = ~128.3 MiB scratch), recompute argmax along the
// taken path in the backward walk. One wave32 per sequence, trans columns
// in VGPRs, alpha broadcast from LDS, async global->LDS double buffering
// via inline CDNA5 asm (global_load_async_to_lds_b128 / s_wait_asynccnt).

#define GO_TAG 1
#define NEGV   (-10000.0f)
#define NB     1024
#define NT     512
#define NK     64
#define NT1    513
#define NWAVE  4      // sequences (waves) per block -> 256 blocks
#define TILE   8      // frames per async staging tile (8*64*4 = 2KB)

// One b128 async copy per lane: 32 lanes x 16B = 512B per instruction.
// VDST = 32-bit LDS byte address, VADDR = 64-bit global address pair.
__device__ __forceinline__ void async_b128(const float* g, float* l) {
#if defined(__HIP_DEVICE_COMPILE__)
  const unsigned lds_off =
      (unsigned)(uintptr_t)(__attribute__((address_space(3))) float*)l;
  const unsigned long long ga = (unsigned long long)(uintptr_t)g;
  asm volatile("global_load_async_to_lds_b128 %0, %1, off"
               :
               : "v"(lds_off), "v"(ga)
               : "memory");
#else
  *(float4*)l = *(const float4*)g;   // host-pass placeholder
#endif
}

#if defined(__HIP_DEVICE_COMPILE__)
#define WAIT_ASYNC(n) asm volatile("s_wait_asynccnt " #n ::: "memory")
#else
#define WAIT_ASYNC(n) ((void)0)
#endif

// Wave-internal ordering for the LDS alpha ping-pong: DS ops execute
// in-order per wave in HW; this just pins the compiler's ordering.
__device__ __forceinline__ void wave_membar() {
  asm volatile("" ::: "memory");
#if defined(__AMDGCN__)
  __builtin_amdgcn_fence(__ATOMIC_ACQ_REL, "wavefront");
#endif
  asm volatile("" ::: "memory");
}

__global__ __launch_bounds__(NWAVE * 32, 1)
void viterbi_fused(const float* __restrict__ unary,   // [B,T,K]
                   const int*   __restrict__ lengths, // [B]
                   const int*   __restrict__ inv_mask,// [K,K]
                   const float* __restrict__ A,       // [K,K]
                   float* __restrict__ out,           // [B,T] tags (as float)
                   float* __restrict__ hist)          // [B,T1,K] alpha history
{
  __shared__ __align__(16) float tr_sh [NK * NK];   // trans[i][j]
  __shared__ __align__(16) float trT_sh[NK * NK];   // trans[j][i] (transposed)
  __shared__ __align__(16) float alpha_sh[NWAVE][2][NK];
  __shared__ __align__(16) float ustage  [NWAVE][2][TILE * NK];

  const int tid = threadIdx.x;

  // ---- Phase 0: trans = log_softmax(A + NEG*inv_mask) over rows ----
  if (tid < NK) {
    const int r = tid;
    float m = -3.0e38f;
    for (int j = 0; j < NK; ++j) {
      const float v = A[r*NK + j] + (inv_mask[r*NK + j] ? NEGV : 0.0f);
      m = fmaxf(m, v);
    }
    float s = 0.0f;
    for (int j = 0; j < NK; ++j) {
      const float v = A[r*NK + j] + (inv_mask[r*NK + j] ? NEGV : 0.0f);
      s += expf(v - m);
    }
    const float lz = m + logf(s);
    for (int j = 0; j < NK; ++j) {
      const float v  = A[r*NK + j] + (inv_mask[r*NK + j] ? NEGV : 0.0f);
      const float tv = v - lz;
      tr_sh [r*NK + j] = tv;
      trT_sh[j*NK + r] = tv;
    }
  }
  __syncthreads();

  const int wave = tid >> 5;
  const int lane = tid & 31;
  const int b    = blockIdx.x * NWAVE + wave;
  const int lraw = lengths[b] + 1;             // +1 for GO frame
  const int len  = (lraw < NT1) ? lraw : NT1;  // steps t valid while t < len

  float*       __restrict__ hb = hist  + (size_t)b * (NT1 * NK);
  const float* __restrict__ ub = unary + (size_t)b * (NT  * NK);
  float*       __restrict__ ob = out   + (size_t)b * NT;

  // Hoist this lane's two trans columns (tags j=lane, j=lane+32) into VGPRs.
  float tc0[NK], tc1[NK];
  #pragma unroll
  for (int i = 0; i < NK; ++i) {
    tc0[i] = tr_sh[i*NK + lane];
    tc1[i] = tr_sh[i*NK + lane + 32];
  }

  // alpha frame 0 = GO frame
  float a0 = (lane == GO_TAG) ? 0.0f : NEGV;
  float a1 = NEGV;
  alpha_sh[wave][0][lane]      = a0;
  alpha_sh[wave][0][lane + 32] = a1;
  hb[lane]      = a0;
  hb[lane + 32] = a1;

  // Prefetch unary tile 0 (frames 0..7)
  { const float* g = ub + lane * 4;
    float*       l = &ustage[wave][0][0] + lane * 4;
    #pragma unroll
    for (int k = 0; k < 4; ++k) async_b128(g + k*128, l + k*128);
  }

  int abuf = 0, curbuf = 0;
  wave_membar();

  // ---- Forward: max-plus recurrence, store alpha history ----
  for (int t = 1; t < len; ++t) {
    const int f = t - 1;                       // unary frame
    if ((f & (TILE - 1)) == 0) {
      const int nb2 = f + TILE;                // next tile base frame
      if (nb2 < NT) {
        const int nbuf = ((f >> 3) + 1) & 1;
        const float* g = ub + nb2*NK + lane * 4;
        float*       l = &ustage[wave][nbuf][0] + lane * 4;
        #pragma unroll
        for (int k = 0; k < 4; ++k) async_b128(g + k*128, l + k*128);
        WAIT_ASYNC(4);                         // current tile complete
      } else {
        WAIT_ASYNC(0);
      }
      curbuf = (f >> 3) & 1;
      wave_membar();
    }

    const float* __restrict__ cur = &alpha_sh[wave][abuf][0];
    // 4 accumulators per tag to break the max dependency chain (dual-issue).
    float p0a = -3.0e38f, p0b = -3.0e38f, p0c = -3.0e38f, p0d = -3.0e38f;
    float p1a = -3.0e38f, p1b = -3.0e38f, p1c = -3.0e38f, p1d = -3.0e38f;
    #pragma unroll
    for (int i = 0; i < NK; i += 4) {
      const float4 av = *(const float4*)(cur + i);  // ds_load_b128 broadcast
      p0a = fmaxf(p0a, av.x + tc0[i + 0]);
      p1a = fmaxf(p1a, av.x + tc1[i + 0]);
      p0b = fmaxf(p0b, av.y + tc0[i + 1]);
      p1b = fmaxf(p1b, av.y + tc1[i + 1]);
      p0c = fmaxf(p0c, av.z + tc0[i + 2]);
      p1c = fmaxf(p1c, av.z + tc1[i + 2]);
      p0d = fmaxf(p0d, av.w + tc0[i + 3]);
      p1d = fmaxf(p1d, av.w + tc1[i + 3]);
    }
    const float best0 = fmaxf(fmaxf(p0a, p0b), fmaxf(p0c, p0d));
    const float best1 = fmaxf(fmaxf(p1a, p1b), fmaxf(p1c, p1d));

    const float* __restrict__ u = &ustage[wave][curbuf][(f & (TILE - 1)) * NK];
    a0 = best0 + u[lane];
    a1 = best1 + u[lane + 32];

    abuf ^= 1;
    alpha_sh[wave][abuf][lane]      = a0;
    alpha_sh[wave][abuf][lane + 32] = a1;
    hb[t*NK + lane]      = a0;                 // alpha history for backward
    hb[t*NK + lane + 32] = a1;
    wave_membar();
  }

  WAIT_ASYNC(0);   // drain leftover prefetch before reusing ustage

  // ---- last = argmax_j(alpha_final), first-occurrence tie rule ----
  float bv = (a1 > a0) ? a1 : a0;
  int   bi = (a1 > a0) ? (lane + 32) : lane;
  #pragma unroll
  for (int off = 16; off >= 1; off >>= 1) {
    const float ov = __shfl_xor(bv, off, 32);
    const int   oi = __shfl_xor(bi, off, 32);
    if (ov > bv || (ov == bv && oi < bi)) { bv = ov; bi = oi; }
  }
  const int last = bi;   // identical in all lanes

  // Frozen tail: out[j] = last for j >= len-2 (coalesced across lanes).
  for (int j = (len - 2) + lane; j < NT; j += 32) ob[j] = (float)last;

#if defined(__AMDGCN__)
  // Make this wave's hist stores visible to the async-DMA reads below.
  __builtin_amdgcn_fence(__ATOMIC_ACQ_REL, "agent");
#endif

  // ---- Backward walk with async-prefetched alpha frames ----
  const int fh  = len - 2;          // highest frame needed (input to step len-1)
  const int ghi = fh >> 3;
  { const float* src = hb + (ghi * TILE) * NK + lane * 4;
    float*       l   = &ustage[wave][ghi & 1][0] + lane * 4;
    #pragma unroll
    for (int k = 0; k < 4; ++k) async_b128(src + k*128, l + k*128);
  }

  int tag = last;
  for (int gg = ghi; gg >= 0; --gg) {
    if (gg > 0) {
      const float* src = hb + ((gg - 1) * TILE) * NK + lane * 4;
      float*       l   = &ustage[wave][(gg - 1) & 1][0] + lane * 4;
      #pragma unroll
      for (int k = 0; k < 4; ++k) async_b128(src + k*128, l + k*128);
      WAIT_ASYNC(4);
    } else {
      WAIT_ASYNC(0);
    }
    wave_membar();

    const float* __restrict__ st = &ustage[wave][gg & 1][0];
    const int ftop = (fh < gg*TILE + TILE - 1) ? fh : (gg*TILE + TILE - 1);
    for (int f = ftop; f >= gg * TILE; --f) {
      const float* __restrict__ af   = st + (f & (TILE - 1)) * NK;
      const float* __restrict__ tcol = &trT_sh[tag * NK];   // conflict-free
      const float c0 = af[lane]      + tcol[lane];
      const float c1 = af[lane + 32] + tcol[lane + 32];
      float v    = (c1 > c0) ? c1 : c0;
      int   bidx = (c1 > c0) ? (lane + 32) : lane;
      #pragma unroll
      for (int off = 16; off >= 1; off >>= 1) {
        const float ov = __shfl_xor(v, off, 32);
        const int   oi = __shfl_xor(bidx, off, 32);
        if (ov > v || (ov == v && oi < bidx)) { v = ov; bidx = oi; }
      }
      tag = bidx;                       // uniform across lanes
      const int t = f + 1;              // this computed tags[t-1]
      if (t >= 2 && lane == 0) ob[t - 2] = (float)tag;
    }
  }
}

extern "C" void kernel_launch(void* const* d_in, const int* in_sizes, int n_in,
                              void* d_out, int out_size, void* d_ws, size_t ws_size,
                              hipStream_t stream) {
  (void)in_sizes; (void)n_in; (void)out_size; (void)ws_size;
  const float* unary    = (const float*)d_in[0];
  const int*   lengths  = (const int*)  d_in[1];
  const int*   inv_mask = (const int*)  d_in[2];
  const float* A        = (const float*)d_in[3];
  float*       out      = (float*)d_out;
  float*       hist     = (float*)d_ws;   // needs NB*NT1*NK*4 ~= 128.3 MiB

  dim3 grid(NB / NWAVE);   // 256 blocks
  dim3 block(NWAVE * 32);  // 4 wave32 per block, one sequence per wave
  hipLaunchKernelGGL(viterbi_fused, grid, block, 0, stream,
                     unary, lengths, inv_mask, A, out, hist);
}